// HybridMemoryMultiFocalPercentClusterUnlabeled_68444598829641
// MI455X (gfx1250) — compile-verified
//
#include <hip/hip_runtime.h>
#include <hip/hip_bf16.h>

// Problem dimensions (fixed by the reference setup)
#define DIMD 256      // feature dim
#define DIMB 256      // batch
#define DIMC 1000     // clusters
#define DIMN 200000   // memory-bank rows
#define MOM  0.5f
#define REPS 1e-12f
#define CTILES 63     // ceil(1000/16)

typedef __attribute__((ext_vector_type(2))) float v2f;
typedef __attribute__((ext_vector_type(8))) float v8f;

// ---------------------------------------------------------------------------
// Pass 0: zero/reset workspace (sums, cnts, touched, rowmap)
// ---------------------------------------------------------------------------
__global__ void zero_ws_kernel(float* __restrict__ sums, int* __restrict__ rowmap,
                               int* __restrict__ cnts, int* __restrict__ touched) {
    int idx    = blockIdx.x * blockDim.x + threadIdx.x;
    int stride = gridDim.x * blockDim.x;
    for (int i = idx; i < DIMC * DIMD; i += stride) sums[i] = 0.0f;
    for (int i = idx; i < DIMN; i += stride) rowmap[i] = -1;
    for (int i = idx; i < DIMC; i += stride) { cnts[i] = 0; touched[i] = 0; }
}

// ---------------------------------------------------------------------------
// Pass 1: momentum update + L2 renorm of the 256 touched rows; build scatter
// maps. One 256-thread block per batch row (one thread per feature element).
// ---------------------------------------------------------------------------
__global__ void update_rows_kernel(const float* __restrict__ inputs,
                                   const float* __restrict__ features,
                                   const int* __restrict__ indexes,
                                   const int* __restrict__ targets,
                                   float* __restrict__ newrows,
                                   int* __restrict__ rowmap,
                                   int* __restrict__ touched) {
    __shared__ float red[DIMD];
    const int b = blockIdx.x;
    const int t = threadIdx.x;
    const int i = indexes[b];

    float v = MOM * features[(size_t)i * DIMD + t] + (1.0f - MOM) * inputs[(size_t)b * DIMD + t];
    red[t] = v * v;
    __syncthreads();
    #pragma unroll
    for (int s = DIMD / 2; s > 0; s >>= 1) {
        if (t < s) red[t] += red[t + s];
        __syncthreads();
    }
    const float inv = 1.0f / sqrtf(red[0] + REPS);
    newrows[(size_t)b * DIMD + t] = v * inv;
    if (t == 0) {
        rowmap[i] = b;           // row -> batch slot (last writer wins on dup)
        touched[targets[b]] = 1; // cluster touched flag
    }
}

// ---------------------------------------------------------------------------
// Pass 2 (the bandwidth-bound pass): one wave per bank row.
//   - select original row or updated row
//   - stream it to the feats output (float4 x2 per lane)
//   - atomically accumulate into per-cluster sums (L2-resident, 1 MB)
//   - lane 0 counts the row into its cluster
// ---------------------------------------------------------------------------
__global__ void bank_scatter_sum_kernel(const float* __restrict__ features,
                                        const float* __restrict__ newrows,
                                        const int* __restrict__ rowmap,
                                        const int* __restrict__ labels,
                                        float* __restrict__ out_feats,
                                        float* __restrict__ sums,
                                        int* __restrict__ cnts) {
    const int wave = (blockIdx.x * blockDim.x + threadIdx.x) >> 5;
    const int lane = threadIdx.x & 31;
    if (wave >= DIMN) return;
    const int row = wave;

    const int b = rowmap[row];
    const float* src = (b >= 0) ? (newrows + (size_t)b * DIMD)
                                : (features + (size_t)row * DIMD);

    const float4* s4 = (const float4*)(src + lane * 8);
    float4 v0 = s4[0];
    float4 v1 = s4[1];

    float4* d4 = (float4*)(out_feats + (size_t)row * DIMD + lane * 8);
    d4[0] = v0;
    d4[1] = v1;

    const int lab = labels[row];
    float* sp = sums + (size_t)lab * DIMD + lane * 8;
    unsafeAtomicAdd(sp + 0, v0.x);
    unsafeAtomicAdd(sp + 1, v0.y);
    unsafeAtomicAdd(sp + 2, v0.z);
    unsafeAtomicAdd(sp + 3, v0.w);
    unsafeAtomicAdd(sp + 4, v1.x);
    unsafeAtomicAdd(sp + 5, v1.y);
    unsafeAtomicAdd(sp + 6, v1.z);
    unsafeAtomicAdd(sp + 7, v1.w);
    if (lane == 0) atomicAdd(&cnts[lab], 1);
}

// ---------------------------------------------------------------------------
// Pass 3: finalize cluster means (touched -> mean, else passthrough)
// ---------------------------------------------------------------------------
__global__ void finalize_means_kernel(const float* __restrict__ cluster_mean,
                                      const float* __restrict__ sums,
                                      const int* __restrict__ cnts,
                                      const int* __restrict__ touched,
                                      float* __restrict__ out_cm) {
    const int idx = blockIdx.x * blockDim.x + threadIdx.x;
    if (idx >= DIMC * DIMD) return;
    const int c = idx / DIMD;
    const float cnt  = (float)cnts[c];
    const float mean = sums[idx] / fmaxf(cnt, 1.0f);
    out_cm[idx] = touched[c] ? mean : cluster_mean[idx];
}

// ---------------------------------------------------------------------------
// Pass 4: logits GEMM  outputs[B,C] = inputs[B,D] * cluster_mean[C,D]^T
// One wave per 16x16 output tile; K-loop of V_WMMA_F32_16X16X4_F32.
// fp32 WMMA layout (ISA 7.12.2):
//   A 16x4 : lane&15 = M, VGPR0/1 = K{0,1} (lanes<16) or K{2,3} (lanes>=16)
//   B 4x16 : lane&15 = N, same K split (B rows of cluster_mean contiguous in D)
//   C/D    : VGPR r -> row r + 8*(lane>=16), col = lane&15
// ---------------------------------------------------------------------------
__global__ void logits_wmma_kernel(const float* __restrict__ A,   // inputs  [B,D]
                                   const float* __restrict__ Bm,  // c_mean  [C,D]
                                   float* __restrict__ out) {     // outputs [B,C]
    const int wave = (blockIdx.x * blockDim.x + threadIdx.x) >> 5;
    const int lane = threadIdx.x & 31;
    const int ntiles = (DIMB / 16) * CTILES;
    if (wave >= ntiles) return;

    const int tM = wave / CTILES;
    const int tN = wave % CTILES;
    const int m0 = tM * 16;
    const int n0 = tN * 16;

    const int lrow = lane & 15;
    const int koff = (lane >> 4) << 1;   // 0 or 2

    const float* aptr = A + (size_t)(m0 + lrow) * DIMD + koff;
    const int   ncol  = n0 + lrow;
    const bool  nval  = (ncol < DIMC);
    const float* bptr = Bm + (size_t)(nval ? ncol : 0) * DIMD + koff;

    v8f acc = {};
    #pragma unroll 8
    for (int k = 0; k < DIMD; k += 4) {
        v2f a = *(const v2f*)(aptr + k);
        v2f b = *(const v2f*)(bptr + k);
        if (!nval) { b.x = 0.0f; b.y = 0.0f; }
        acc = __builtin_amdgcn_wmma_f32_16x16x4_f32(
            /*neg_a=*/false, a, /*neg_b=*/false, b,
            /*c_mod=*/(short)0, acc, /*reuse_a=*/false, /*reuse_b=*/false);
    }

    const int row = m0 + ((lane >> 4) << 3);
    const int col = n0 + lrow;
    if (col < DIMC) {
        #pragma unroll
        for (int r = 0; r < 8; ++r)
            out[(size_t)(row + r) * DIMC + col] = acc[r];
    }
}

// ---------------------------------------------------------------------------
extern "C" void kernel_launch(void* const* d_in, const int* in_sizes, int n_in,
                              void* d_out, int out_size, void* d_ws, size_t ws_size,
                              hipStream_t stream) {
    const float* inputs       = (const float*)d_in[0]; // [256,256]
    const float* features     = (const float*)d_in[1]; // [200000,256]
    const float* cluster_mean = (const float*)d_in[2]; // [1000,256]
    const int*   indexes      = (const int*)d_in[3];   // [256]
    const int*   labels       = (const int*)d_in[4];   // [200000]
    const int*   targets      = (const int*)d_in[5];   // [256]

    // Output layout (flat concat, return order): outputs, feats, new_cluster_mean
    float* out_logits = (float*)d_out;                           // 256*1000
    float* out_feats  = out_logits + (size_t)DIMB * DIMC;        // 200000*256
    float* out_cm     = out_feats + (size_t)DIMN * DIMD;         // 1000*256

    // Workspace layout
    float* sums    = (float*)d_ws;                               // 1000*256 f32
    float* newrows = sums + (size_t)DIMC * DIMD;                 // 256*256 f32
    int*   rowmap  = (int*)(newrows + (size_t)DIMB * DIMD);      // 200000 i32
    int*   cnts    = rowmap + DIMN;                              // 1000 i32
    int*   touched = cnts + DIMC;                                // 1000 i32

    // 0) reset workspace
    zero_ws_kernel<<<1024, 256, 0, stream>>>(sums, rowmap, cnts, touched);

    // 1) momentum-update + renorm the 256 touched rows; build maps
    update_rows_kernel<<<DIMB, DIMD, 0, stream>>>(inputs, features, indexes,
                                                  targets, newrows, rowmap, touched);

    // 2) fused bank copy + scatter + per-cluster segment-sum (one wave / row)
    bank_scatter_sum_kernel<<<DIMN / 8, 256, 0, stream>>>(features, newrows, rowmap,
                                                          labels, out_feats, sums, cnts);

    // 3) finalize cluster means
    finalize_means_kernel<<<(DIMC * DIMD + 255) / 256, 256, 0, stream>>>(
        cluster_mean, sums, cnts, touched, out_cm);

    // 4) logits GEMM via fp32 WMMA (independent of the bank update)
    const int ntiles = (DIMB / 16) * CTILES;        // 1008 waves
    logits_wmma_kernel<<<(ntiles + 7) / 8, 256, 0, stream>>>(inputs, cluster_mean,
                                                             out_logits);
}